// One_Neighbor_Aggregation_14568529068220
// MI455X (gfx1250) — compile-verified
//
#include <hip/hip_runtime.h>

#define N_NODES 50000
#define D_DIM   128
#define E_EDGES 640000

typedef __attribute__((ext_vector_type(2))) float v2f;
typedef __attribute__((ext_vector_type(8))) float v8f;

// ---------------------------------------------------------------------------
// Pass 0: zero the workspace (agg[N*128] | esum[N] | deg[N])
// ---------------------------------------------------------------------------
__global__ void zero_ws_kernel(float* __restrict__ p, int n) {
  int i = blockIdx.x * blockDim.x + threadIdx.x;
  if (i < n) p[i] = 0.0f;
}

// ---------------------------------------------------------------------------
// Pass 1: edge phase. One wave32 per edge; each lane handles 4 consecutive
// feature dims (float4 load + 4 native f32 atomics into L2-resident agg).
//   agg[dst]  += e * feat[src]
//   esum[dst] += e
//   deg[dst]  += 1
// ---------------------------------------------------------------------------
__global__ void edge_scatter_kernel(const float* __restrict__ feat,
                                    const int*   __restrict__ esrc,
                                    const int*   __restrict__ edst,
                                    const float* __restrict__ ee,
                                    float* __restrict__ agg,
                                    float* __restrict__ esum,
                                    float* __restrict__ deg,
                                    int E) {
  int gid  = blockIdx.x * blockDim.x + threadIdx.x;
  int e    = gid >> 5;   // wave-uniform edge index -> scalar loads of edge data
  int lane = gid & 31;
  if (e >= E) return;

  int   src = esrc[e];
  int   dst = edst[e];
  float w   = ee[e];

  const float4* f4 = (const float4*)(feat + (size_t)src * D_DIM);
  float4 v = f4[lane];

  float* a = agg + (size_t)dst * D_DIM + lane * 4;
  unsafeAtomicAdd(a + 0, w * v.x);
  unsafeAtomicAdd(a + 1, w * v.y);
  unsafeAtomicAdd(a + 2, w * v.z);
  unsafeAtomicAdd(a + 3, w * v.w);

  if (lane == 0) {
    unsafeAtomicAdd(esum + dst, w);
    unsafeAtomicAdd(deg  + dst, 1.0f);
  }
}

// ---------------------------------------------------------------------------
// Pass 2: fused GEMM + epilogue via V_WMMA_F32_16X16X4_F32 (pure f32 path).
//   pre  = agg @ W^T + esum*b        (tile: 16 nodes x 16 out-dims, K=128)
//   out  = relu(pre / max(deg,1)) + alpha*feat
// One wave per 16x16 tile; EXEC all-ones (50000 = 16*3125 exactly).
// ---------------------------------------------------------------------------
__global__ void gemm_epilogue_kernel(const float* __restrict__ agg,
                                     const float* __restrict__ esum,
                                     const float* __restrict__ deg,
                                     const float* __restrict__ Wm,
                                     const float* __restrict__ bias,
                                     const float* __restrict__ feat,
                                     const float* __restrict__ alpha,
                                     float* __restrict__ out) {
  int lane = threadIdx.x;      // 0..31
  int half = lane >> 4;        // 0 or 1
  int lid  = lane & 15;

  int m0 = blockIdx.x * 16;    // node tile base
  int n0 = blockIdx.y * 16;    // output-dim tile base

  int m = m0 + lid;            // A-matrix row owned by this lane
  int j = n0 + lid;            // B-matrix column owned by this lane

  // A 16x4 f32 layout: lanes 0-15 hold K={k0,k0+1}, lanes 16-31 K={k0+2,k0+3}
  const float* arow = agg + (size_t)m * D_DIM + 2 * half;
  // B = W^T, i.e. B[k][j] = W[j][k]; mirrored K layout
  const float* wrow = Wm  + (size_t)j * D_DIM + 2 * half;

  v8f c = {};
  #pragma unroll
  for (int k0 = 0; k0 < D_DIM; k0 += 4) {
    v2f a = *(const v2f*)(arow + k0);
    v2f b = *(const v2f*)(wrow + k0);
    // 8 args: (neg_a, A, neg_b, B, c_mod, C, reuse_a, reuse_b)
    c = __builtin_amdgcn_wmma_f32_16x16x4_f32(
        false, a, false, b, (short)0, c, false, false);
  }

  // C/D layout: VGPR v, lanes 0-15 -> row m0+v, lanes 16-31 -> row m0+v+8
  #pragma unroll
  for (int v = 0; v < 8; ++v) {
    int n = m0 + v + 8 * half;
    float val = c[v] + esum[n] * bias[j];
    float dg  = deg[n];
    dg  = dg > 1.0f ? dg : 1.0f;
    val = val / dg;
    val = val > 0.0f ? val : 0.0f;             // relu
    size_t idx = (size_t)n * D_DIM + j;
    out[idx] = val + alpha[n] * feat[idx];     // residual
  }
}

// ---------------------------------------------------------------------------
extern "C" void kernel_launch(void* const* d_in, const int* in_sizes, int n_in,
                              void* d_out, int out_size, void* d_ws, size_t ws_size,
                              hipStream_t stream) {
  const float* feat  = (const float*)d_in[0];
  const float* alpha = (const float*)d_in[1];
  const int*   esrc  = (const int*)  d_in[2];
  const int*   edst  = (const int*)  d_in[3];
  const float* ee    = (const float*)d_in[4];
  const float* Wm    = (const float*)d_in[5];
  const float* bias  = (const float*)d_in[6];
  float*       out   = (float*)d_out;

  float* agg  = (float*)d_ws;                    // N*128 floats
  float* esum = agg  + (size_t)N_NODES * D_DIM;  // N floats
  float* deg  = esum + N_NODES;                  // N floats

  // Pass 0: zero workspace
  int total = N_NODES * (D_DIM + 2);
  zero_ws_kernel<<<(total + 255) / 256, 256, 0, stream>>>(agg, total);

  // Pass 1: edge weighted scatter (one wave32 per edge)
  long long ethreads = (long long)E_EDGES * 32;
  edge_scatter_kernel<<<(unsigned)((ethreads + 255) / 256), 256, 0, stream>>>(
      feat, esrc, edst, ee, agg, esum, deg, E_EDGES);

  // Pass 2: WMMA GEMM + fused epilogue (one wave per 16x16 tile)
  dim3 grid(N_NODES / 16, D_DIM / 16);  // 3125 x 8
  gemm_epilogue_kernel<<<grid, 32, 0, stream>>>(
      agg, esum, deg, Wm, bias, feat, alpha, out);
}